// LateralInhibition_calculate_84670985273439
// MI455X (gfx1250) — compile-verified
//
#include <hip/hip_runtime.h>
#include <math.h>

typedef float v2f __attribute__((ext_vector_type(2)));
typedef float v8f __attribute__((ext_vector_type(8)));

constexpr int   BATCH  = 16;
constexpr int   H      = 512;
constexpr int   W      = 512;
constexpr int   NPIX   = BATCH * H * W;          // 4,194,304
constexpr int   L      = 7;
constexpr int   RAD    = 3;
constexpr float A_COEF = 0.1f;
constexpr float B_COEF = 0.9f;

constexpr int TW = 64, TH = 32;                  // pixels per block (2048)
constexpr int TROWS = TH + 2 * RAD;              // 38
constexpr int TCOLS = TW + 2 * RAD;              // 70
constexpr int LS    = 72;                        // LDS row stride: quad bases stay 16B aligned
constexpr int NBLK  = (W / TW) * (H / TH) * BATCH; // 2048 blocks in pass 1

// ---------------------------------------------------------------------------
// Pass 1: 64x32 tile + 3-halo in LDS. Each thread owns two pixel QUADS
// (column aligned to 4): per stencil row that is 2x ds_load_b128 + 1x
// ds_load_b64 feeding packed-fp32 relu/fma for 4 pixels. avg/diff go straight
// to their d_out sections via float4 stores; per-block sum(sup^2) partial is
// reduced wave-wide with fp32 WMMA (16x16x4, B = ones) then one LDS combine.
// ---------------------------------------------------------------------------
__global__ __launch_bounds__(256)
void li_pass1(const float* __restrict__ x, const float* __restrict__ kern,
              float* __restrict__ out, float* __restrict__ partials)
{
    __shared__ __align__(16) float tile[TROWS * LS];
    __shared__ float wred[16];

    const int tid = threadIdx.x;
    const int gx0 = blockIdx.x * TW;
    const int gy0 = blockIdx.y * TH;
    const float* img = x + (size_t)blockIdx.z * H * W;

    // Cooperative zero-padded halo load (38 x 70).
    for (int i = tid; i < TROWS * TCOLS; i += 256) {
        int r = i / TCOLS, c = i - r * TCOLS;
        int yy = gy0 + r - RAD, xx = gx0 + c - RAD;
        float v = 0.0f;
        if ((unsigned)yy < (unsigned)H && (unsigned)xx < (unsigned)W)
            v = img[yy * W + xx];
        tile[r * LS + c] = v;
    }

    // Weights are uniform across the block -> scalar loads / SGPR operands.
    float wk[L * L];
#pragma unroll
    for (int i = 0; i < L * L; i++) wk[i] = kern[i];

    __syncthreads();

    const size_t base = (size_t)blockIdx.z * H * W;
    float local_ss = 0.0f;

#pragma unroll
    for (int k = 0; k < 2; k++) {
        const int q  = tid + k * 256;     // quad index 0..511
        const int qr = q >> 4;            // tile row 0..31
        const int qc = (q & 15) * 4;      // col 0,4,...,60 (16B aligned)

        const float* ctr = &tile[(qr + RAD) * LS + qc + RAD];
        const float c0 = ctr[0], c1 = ctr[1], c2 = ctr[2], c3 = ctr[3];
        const v2f cc0 = {c0, c0}, cc1 = {c1, c1}, cc2 = {c2, c2}, cc3 = {c3, c3};
        const v2f z = {0.0f, 0.0f};

        v2f acc0 = z, acc1 = z, acc2 = z, acc3 = z;
        float sum0 = 0.0f, sum1 = 0.0f, sum2 = 0.0f, sum3 = 0.0f;

#pragma unroll
        for (int ty = 0; ty < L; ty++) {
            const float* rowp = &tile[(qr + ty) * LS + qc];
            const float4 A4 = *(const float4*)rowp;        // taps 0..3
            const float4 B4 = *(const float4*)(rowp + 4);  // taps 4..7
            const v2f    C2 = *(const v2f*)(rowp + 8);     // taps 8..9

            const v2f p0 = {A4.x, A4.y};
            const v2f p1 = {A4.z, A4.w};
            const v2f p2 = {B4.x, B4.y};
            const v2f p3 = {B4.z, B4.w};
            const v2f p4 = C2;

            // Box sums: hh = taps2..7; windows are 7-wide starting at j.
            v2f hs = (p1 + p2) + p3;
            float hh = hs.x + hs.y;
            sum0 += (hh - B4.w) + (A4.x + A4.y);   // t0..t6
            sum1 += hh + A4.y;                     // t1..t7
            sum2 += hh + C2.x;                     // t2..t8
            sum3 += (hh - A4.z) + (C2.x + C2.y);   // t3..t9

            // Weight pairs (uniform -> SGPR pairs for v_pk_fma_f32).
            const v2f w01 = {wk[ty * 7 + 0], wk[ty * 7 + 1]};
            const v2f w23 = {wk[ty * 7 + 2], wk[ty * 7 + 3]};
            const v2f w45 = {wk[ty * 7 + 4], wk[ty * 7 + 5]};
            const v2f w6z = {wk[ty * 7 + 6], 0.0f};
            const v2f wz0 = {0.0f, wk[ty * 7 + 0]};
            const v2f w12 = {wk[ty * 7 + 1], wk[ty * 7 + 2]};
            const v2f w34 = {wk[ty * 7 + 3], wk[ty * 7 + 4]};
            const v2f w56 = {wk[ty * 7 + 5], wk[ty * 7 + 6]};

            // pixel 0: taps 0..6  (pairs p0..p3, pattern A)
            acc0 = __builtin_elementwise_fma(__builtin_elementwise_max(p0 - cc0, z), w01, acc0);
            acc0 = __builtin_elementwise_fma(__builtin_elementwise_max(p1 - cc0, z), w23, acc0);
            acc0 = __builtin_elementwise_fma(__builtin_elementwise_max(p2 - cc0, z), w45, acc0);
            acc0 = __builtin_elementwise_fma(__builtin_elementwise_max(p3 - cc0, z), w6z, acc0);
            // pixel 1: taps 1..7  (pairs p0..p3, pattern B)
            acc1 = __builtin_elementwise_fma(__builtin_elementwise_max(p0 - cc1, z), wz0, acc1);
            acc1 = __builtin_elementwise_fma(__builtin_elementwise_max(p1 - cc1, z), w12, acc1);
            acc1 = __builtin_elementwise_fma(__builtin_elementwise_max(p2 - cc1, z), w34, acc1);
            acc1 = __builtin_elementwise_fma(__builtin_elementwise_max(p3 - cc1, z), w56, acc1);
            // pixel 2: taps 2..8  (pairs p1..p4, pattern A)
            acc2 = __builtin_elementwise_fma(__builtin_elementwise_max(p1 - cc2, z), w01, acc2);
            acc2 = __builtin_elementwise_fma(__builtin_elementwise_max(p2 - cc2, z), w23, acc2);
            acc2 = __builtin_elementwise_fma(__builtin_elementwise_max(p3 - cc2, z), w45, acc2);
            acc2 = __builtin_elementwise_fma(__builtin_elementwise_max(p4 - cc2, z), w6z, acc2);
            // pixel 3: taps 3..9  (pairs p1..p4, pattern B)
            acc3 = __builtin_elementwise_fma(__builtin_elementwise_max(p1 - cc3, z), wz0, acc3);
            acc3 = __builtin_elementwise_fma(__builtin_elementwise_max(p2 - cc3, z), w12, acc3);
            acc3 = __builtin_elementwise_fma(__builtin_elementwise_max(p3 - cc3, z), w34, acc3);
            acc3 = __builtin_elementwise_fma(__builtin_elementwise_max(p4 - cc3, z), w56, acc3);
        }

        const float diff0 = acc0.x + acc0.y;
        const float diff1 = acc1.x + acc1.y;
        const float diff2 = acc2.x + acc2.y;
        const float diff3 = acc3.x + acc3.y;
        const float avg0 = __expf(-sum0 * (1.0f / 49.0f));
        const float avg1 = __expf(-sum1 * (1.0f / 49.0f));
        const float avg2 = __expf(-sum2 * (1.0f / 49.0f));
        const float avg3 = __expf(-sum3 * (1.0f / 49.0f));
        const float sup0 = A_COEF * avg0 + B_COEF * diff0;
        const float sup1 = A_COEF * avg1 + B_COEF * diff1;
        const float sup2 = A_COEF * avg2 + B_COEF * diff2;
        const float sup3 = A_COEF * avg3 + B_COEF * diff3;

        const size_t idx = base + (size_t)(gy0 + qr) * W + (gx0 + qc); // mult of 4
        const float4 av = {avg0, avg1, avg2, avg3};
        const float4 dv = {diff0, diff1, diff2, diff3};
        *(float4*)&out[(size_t)NPIX + idx]     = av;   // average_term section
        *(float4*)&out[(size_t)2 * NPIX + idx] = dv;   // differential_term section

        local_ss += (sup0 * sup0 + sup1 * sup1) + (sup2 * sup2 + sup3 * sup3);
    }

    // ---- wave-level reduction via V_WMMA_F32_16X16X4_F32 (exact fp32) ----
    // A[lane] = (local_ss, 0); B = all-ones => D[m][n] = rowsum_m. Summing the
    // 8 C/D VGPRs yields the half-wave total in every lane of that half.
    v2f a     = {local_ss, 0.0f};
    v2f bones = {1.0f, 1.0f};
    v8f cz    = {0.0f, 0.0f, 0.0f, 0.0f, 0.0f, 0.0f, 0.0f, 0.0f};
    v8f d = __builtin_amdgcn_wmma_f32_16x16x4_f32(
        /*neg_a=*/false, a, /*neg_b=*/false, bones,
        /*c_mod=*/(short)0, cz, /*reuse_a=*/false, /*reuse_b=*/false);
    float halfsum = ((d[0] + d[1]) + (d[2] + d[3])) +
                    ((d[4] + d[5]) + (d[6] + d[7]));

    const int lane = tid & 31;
    const int wid  = tid >> 5;
    if (lane == 0)  wred[wid * 2 + 0] = halfsum;  // rows 0..7
    if (lane == 16) wred[wid * 2 + 1] = halfsum;  // rows 8..15
    __syncthreads();

    if (tid == 0) {
        float s = 0.0f;
#pragma unroll
        for (int i = 0; i < 16; i++) s += wred[i];
        partials[(blockIdx.z * gridDim.y + blockIdx.y) * gridDim.x + blockIdx.x] = s;
    }
}

// ---------------------------------------------------------------------------
// Pass 2: deterministic reduction of block partials -> norm.
// ---------------------------------------------------------------------------
__global__ __launch_bounds__(256)
void li_reduce(const float* __restrict__ partials, float* __restrict__ norm_out)
{
    __shared__ float sh[256];
    float s = 0.0f;
    for (int i = threadIdx.x; i < NBLK; i += 256) s += partials[i];
    sh[threadIdx.x] = s;
    __syncthreads();
#pragma unroll
    for (int off = 128; off > 0; off >>= 1) {
        if (threadIdx.x < off) sh[threadIdx.x] += sh[threadIdx.x + off];
        __syncthreads();
    }
    if (threadIdx.x == 0) norm_out[0] = sqrtf(sh[0]);
}

// ---------------------------------------------------------------------------
// Pass 3: mask = (x > sup/norm) == (x*norm > sup), sup rebuilt from the
// avg/diff sections already in d_out. Fully vectorized (float4).
// ---------------------------------------------------------------------------
__global__ __launch_bounds__(256)
void li_mask(const float* __restrict__ x, const float* __restrict__ norm_p,
             float* __restrict__ out)
{
    const float nrm = norm_p[0];
    const int i = blockIdx.x * 256 + threadIdx.x;   // over NPIX/4
    const float4 xv = ((const float4*)x)[i];
    const float4 av = ((const float4*)(out + (size_t)NPIX))[i];
    const float4 dv = ((const float4*)(out + (size_t)2 * NPIX))[i];
    float4 m;
    m.x = (xv.x * nrm > A_COEF * av.x + B_COEF * dv.x) ? 1.0f : 0.0f;
    m.y = (xv.y * nrm > A_COEF * av.y + B_COEF * dv.y) ? 1.0f : 0.0f;
    m.z = (xv.z * nrm > A_COEF * av.z + B_COEF * dv.z) ? 1.0f : 0.0f;
    m.w = (xv.w * nrm > A_COEF * av.w + B_COEF * dv.w) ? 1.0f : 0.0f;
    ((float4*)out)[i] = m;
}

// ---------------------------------------------------------------------------
extern "C" void kernel_launch(void* const* d_in, const int* in_sizes, int n_in,
                              void* d_out, int out_size, void* d_ws, size_t ws_size,
                              hipStream_t stream)
{
    const float* x    = (const float*)d_in[0];   // [16,1,512,512]
    const float* kern = (const float*)d_in[1];   // [49]
    float* out        = (float*)d_out;           // [mask | avg | diff], 3*NPIX
    float* partials   = (float*)d_ws;            // NBLK floats
    float* norm_p     = partials + NBLK;         // 1 float

    dim3 g1(W / TW, H / TH, BATCH);              // 8 x 16 x 16 = 2048 blocks
    li_pass1<<<g1, 256, 0, stream>>>(x, kern, out, partials);
    li_reduce<<<1, 256, 0, stream>>>(partials, norm_p);
    li_mask<<<NPIX / 4 / 256, 256, 0, stream>>>(x, norm_p, out);
}